// Loss_add_81398220193792
// MI455X (gfx1250) — compile-verified
//
#include <hip/hip_runtime.h>
#include <math.h>

typedef float v2f __attribute__((ext_vector_type(2)));
typedef float v8f __attribute__((ext_vector_type(8)));

#define NP 2048
#define BS 32
#define CHUNKS 8            // row-chunks per batch -> grid = 32*8 = 256 blocks
#define ROWS_PER_BLOCK 256  // 256 threads, 8 waves, 2 n-tiles (of 16 rows) per wave
#define TILES_PER_WAVE 2

__global__ __launch_bounds__(64)
void init_ws_kernel(float* ws) {
    ws[threadIdx.x] = 0.0f;   // ws[0..31]=sum||tf-target||, ws[32..63]=sum sqrt(min d2)
}

__global__ __launch_bounds__(256)
void chamfer_wmma_kernel(const float* __restrict__ pred_r,
                         const float* __restrict__ pred_t,
                         const float* __restrict__ target,
                         const float* __restrict__ model,
                         float* __restrict__ ws)
{
    // Target points in WMMA-B layout: per point [x, y, z, |t|^2]  (32 KB)
    __shared__ float sTgt[NP * 4];

    const int b     = blockIdx.x >> 3;   // batch
    const int chunk = blockIdx.x & 7;    // which 256-row slice of this batch
    const int tid   = threadIdx.x;
    const int lane  = tid & 31;
    const int wave  = tid >> 5;
    const int half  = lane >> 4;         // 0: lanes 0-15, 1: lanes 16-31
    const int col   = lane & 15;

    // --- quaternion (wxyz) -> rotation, translation (uniform per block) ---
    float qw = pred_r[b * 4 + 0], qx = pred_r[b * 4 + 1];
    float qy = pred_r[b * 4 + 2], qz = pred_r[b * 4 + 3];
    float qn = rsqrtf(qw * qw + qx * qx + qy * qy + qz * qz);
    qw *= qn; qx *= qn; qy *= qn; qz *= qn;
    // R[d][e], tf[e] = sum_d m[d]*R[d][e] + t[e]
    const float R00 = 1.f - 2.f * (qy * qy + qz * qz);
    const float R01 = 2.f * (qx * qy - qz * qw);
    const float R02 = 2.f * (qx * qz + qy * qw);
    const float R10 = 2.f * (qx * qy + qz * qw);
    const float R11 = 1.f - 2.f * (qx * qx + qz * qz);
    const float R12 = 2.f * (qy * qz - qx * qw);
    const float R20 = 2.f * (qx * qz - qy * qw);
    const float R21 = 2.f * (qy * qz + qx * qw);
    const float R22 = 1.f - 2.f * (qx * qx + qy * qy);
    const float tx = pred_t[b * 3 + 0], ty = pred_t[b * 3 + 1], tz = pred_t[b * 3 + 2];

    const float* tb = target + (size_t)b * NP * 3;
    const float* mb = model  + (size_t)b * NP * 3;

    // --- stage target tiles into LDS (B-matrix layout) ---
    for (int p = tid; p < NP; p += 256) {
        float x = tb[p * 3 + 0], y = tb[p * 3 + 1], z = tb[p * 3 + 2];
        sTgt[p * 4 + 0] = x;
        sTgt[p * 4 + 1] = y;
        sTgt[p * 4 + 2] = z;
        sTgt[p * 4 + 3] = x * x + y * y + z * z;
    }

    // --- elementwise branch: dis = mean ||tf_n - target_n|| (this block's rows) ---
    {
        int row = chunk * ROWS_PER_BLOCK + tid;
        float mx = mb[row * 3 + 0], my = mb[row * 3 + 1], mz = mb[row * 3 + 2];
        float fx = mx * R00 + my * R10 + mz * R20 + tx;
        float fy = mx * R01 + my * R11 + mz * R21 + ty;
        float fz = mx * R02 + my * R12 + mz * R22 + tz;
        float dx = fx - tb[row * 3 + 0];
        float dy = fy - tb[row * 3 + 1];
        float dz = fz - tb[row * 3 + 2];
        float d = sqrtf(dx * dx + dy * dy + dz * dz);
        #pragma unroll
        for (int off = 16; off > 0; off >>= 1) d += __shfl_xor(d, off, 32);
        if (lane == 0) atomicAdd(&ws[b], d);
    }

    __syncthreads();

    // --- nearest-neighbor branch via V_WMMA_F32_16X16X4_F32 ---
    // score[n,m] = |t_m|^2 - 2*tf_n.t_m  ;  min d2 = |tf_n|^2 + min_m score
    float dis2_acc = 0.0f;

    for (int it = 0; it < TILES_PER_WAVE; ++it) {
        const int row0 = chunk * ROWS_PER_BLOCK + (wave * TILES_PER_WAVE + it) * 16;
        const int row  = row0 + col;   // both half-groups cover the same 16 rows

        float mx = mb[row * 3 + 0], my = mb[row * 3 + 1], mz = mb[row * 3 + 2];
        float fx = mx * R00 + my * R10 + mz * R20 + tx;
        float fy = mx * R01 + my * R11 + mz * R21 + ty;
        float fz = mx * R02 + my * R12 + mz * R22 + tz;
        float tfsq = fx * fx + fy * fy + fz * fz;

        // A (16x4): lanes 0-15 hold K={0,1}, lanes 16-31 hold K={2,3}
        v2f a;
        if (half == 0) { a.x = -2.f * fx; a.y = -2.f * fy; }
        else           { a.x = -2.f * fz; a.y = 1.0f;      }

        float minv[8];
        #pragma unroll
        for (int k = 0; k < 8; ++k) minv[k] = 3.4e38f;

        #pragma unroll 4
        for (int mt = 0; mt < NP / 16; ++mt) {
            // B (4x16): lane -> target point (mt*16+col), K = half*2 + {0,1} -> ds_load_b64
            const v2f bv = *(const v2f*)&sTgt[(mt * 16 + col) * 4 + half * 2];
            v8f c = {0.f, 0.f, 0.f, 0.f, 0.f, 0.f, 0.f, 0.f};
            v8f d = __builtin_amdgcn_wmma_f32_16x16x4_f32(
                        false, a, false, bv, (short)0, c, false, false);
            #pragma unroll
            for (int k = 0; k < 8; ++k) minv[k] = fminf(minv[k], d[k]);
        }

        // reduce min across the 16 columns (lanes within each half-group)
        #pragma unroll
        for (int k = 0; k < 8; ++k) {
            float v = minv[k];
            v = fminf(v, __shfl_xor(v, 1, 16));
            v = fminf(v, __shfl_xor(v, 2, 16));
            v = fminf(v, __shfl_xor(v, 4, 16));
            v = fminf(v, __shfl_xor(v, 8, 16));
            minv[k] = v;
        }

        // D layout: VGPR k, lane l -> row M = k + 8*(l/16). Lane l==k has tfsq of
        // row k (half 0), lane l==k+24 has tfsq of row k+8 (half 1).
        #pragma unroll
        for (int k = 0; k < 8; ++k) {
            if (lane == k || lane == k + 24) {
                dis2_acc += sqrtf(fmaxf(tfsq + minv[k], 0.0f));
            }
        }
    }

    #pragma unroll
    for (int off = 16; off > 0; off >>= 1) dis2_acc += __shfl_xor(dis2_acc, off, 32);
    if (lane == 0) atomicAdd(&ws[BS + b], dis2_acc);
}

__global__ __launch_bounds__(32)
void finalize_kernel(const float* __restrict__ ws,
                     const int* __restrict__ idx,
                     float* __restrict__ out)
{
    int b = threadIdx.x;
    if (b >= BS) return;
    float dis = ws[b] * (1.0f / NP);
    int iv = idx[b];
    bool sym = (iv == 0) || (iv == 2) || (iv == 5);
    out[b]      = dis;
    out[BS + b] = sym ? (ws[BS + b] * (1.0f / NP)) : dis;
}

extern "C" void kernel_launch(void* const* d_in, const int* in_sizes, int n_in,
                              void* d_out, int out_size, void* d_ws, size_t ws_size,
                              hipStream_t stream) {
    const float* pred_r = (const float*)d_in[0];   // [32,4]
    const float* pred_t = (const float*)d_in[1];   // [32,3]
    const float* target = (const float*)d_in[2];   // [32,2048,3]
    const float* model  = (const float*)d_in[3];   // [32,2048,3]
    const int*   idx    = (const int*)d_in[4];     // [32,1]
    float* ws  = (float*)d_ws;                     // 64 accumulators
    float* out = (float*)d_out;                    // [2,32]

    init_ws_kernel<<<1, 64, 0, stream>>>(ws);
    chamfer_wmma_kernel<<<BS * CHUNKS, 256, 0, stream>>>(pred_r, pred_t, target, model, ws);
    finalize_kernel<<<1, 32, 0, stream>>>(ws, idx, out);
}